// NoisyTopkRouter_39195871543620
// MI455X (gfx1250) — compile-verified
//
#include <hip/hip_runtime.h>

// NoisyTopkRouter for MI455X (gfx1250, wave32, WMMA).
// B=4, S=4096, D=2048, E=64, TOP_K=2  ->  16384 tokens.
//
// v2: 64 tokens per workgroup, 4 accumulators per wave -> each weight (B)
// fragment is reused across 4 WMMAs, cutting L2 weight traffic 4x (1GB->256MB)
// so it no longer dominates the 134MB HBM stream of mh_output.

typedef __attribute__((ext_vector_type(2))) float v2f;
typedef __attribute__((ext_vector_type(8))) float v8f;

#define N_TOK   16384
#define DDIM    2048
#define EDIM    64
#define ROWS    64            // tokens per workgroup (4 sub-tiles of 16)
#define RT      4             // row sub-tiles per wave
#define KC      64            // K chunk depth staged in LDS
#define NCHUNK  (DDIM / KC)   // 32
#define ASTR    68            // LDS A row stride (floats): 272B rows -> 16B aligned,
                              // conflict-free 2-dword reads across 32 lanes
#define RSTR    72            // LDS result row stride (floats)
#define AELEMS  (ROWS * ASTR) // one A buffer
#define RELEMS  (ROWS * RSTR) // one result matrix

// Aliased LDS: A double-buffer (2*ROWS*ASTR = 8704 floats) lives in the same
// space as the result tiles (2*ROWS*RSTR = 9216 floats). A is dead after the
// K loop (guarded by its trailing barrier).
#define SMEM_FLOATS (2 * RELEMS)

__global__ __launch_bounds__(256)
void noisy_topk_router_kernel(const float* __restrict__ X,      // [N_TOK, D]
                              const float* __restrict__ Wr,     // [D, E]
                              const float* __restrict__ br,     // [E]
                              const float* __restrict__ Wn,     // [D, E]
                              const float* __restrict__ bn,     // [E]
                              const float* __restrict__ noise,  // [N_TOK, E]
                              float* __restrict__ probs,        // [N_TOK, E]
                              int*   __restrict__ indices)      // [N_TOK, 2]
{
    __shared__ __align__(16) float smem[SMEM_FLOATS];
    __shared__ float tp1[ROWS], tp2[ROWS];
    __shared__ int   tix1[ROWS], tix2[ROWS];

    float* __restrict__ aBuf = smem;   // [2][ROWS*ASTR] during GEMM
    float* __restrict__ resb = smem;   // [2][ROWS*RSTR] after GEMM (aliased)

    const int t    = threadIdx.x;
    const int lane = t & 31;
    const int wave = t >> 5;                 // 0..7
    const int tok0 = blockIdx.x * ROWS;

    const int   mat = wave >> 2;             // 0 = route, 1 = noise
    const int   e0  = (wave & 3) * 16;       // expert column tile base
    const float* __restrict__ W = mat ? Wn : Wr;

    // --- staging geometry: 256 threads x 4 float4 = 64 rows x 64 floats ---
    const int sm = t >> 4;                   // base row 0..15 (rows sm, sm+16, sm+32, sm+48)
    const int sf = t & 15;                   // float4 slot 0..15

    // stage chunk 0
    #pragma unroll
    for (int j = 0; j < RT; ++j) {
        const float4 v = *(const float4*)(X + (size_t)(tok0 + sm + 16 * j) * DDIM + sf * 4);
        *(float4*)&aBuf[(sm + 16 * j) * ASTR + sf * 4] = v;
    }
    __syncthreads();

    // --- WMMA fragment geometry (32-bit A: 16x4, B: 4x16, C/D: 16x16) ---
    const int mrow = lane & 15;              // A row (M) and B column (N)
    const int koff = (lane >> 4) * 2;        // lanes 0-15: K{0,1}; lanes 16-31: K{2,3}

    v8f acc[RT];
    #pragma unroll
    for (int rt = 0; rt < RT; ++rt) acc[rt] = (v8f){};

    for (int c = 0; c < NCHUNK; ++c) {
        const int kBase = c * KC;
        const int buf   = c & 1;

        // prefetch next A chunk into registers (overlaps with WMMA work)
        float4 stage[RT];
        if (c + 1 < NCHUNK) {
            #pragma unroll
            for (int j = 0; j < RT; ++j) {
                const float* src = X + (size_t)(tok0 + sm + 16 * j) * DDIM + (kBase + KC) + sf * 4;
                stage[j] = *(const float4*)src;
                if (c + 2 < NCHUNK)          // hint following chunk into L2
                    __builtin_prefetch(src + KC, 0, 1);
            }
        }

        // 16 K-steps; each B fragment feeds 4 independent accumulator chains
        #pragma unroll
        for (int kk = 0; kk < KC; kk += 4) {
            const int krow = kBase + kk + koff;
            v2f b;
            b.x = W[(size_t)(krow + 0) * EDIM + e0 + mrow];
            b.y = W[(size_t)(krow + 1) * EDIM + e0 + mrow];
            #pragma unroll
            for (int rt = 0; rt < RT; ++rt) {
                const v2f a = *(const v2f*)&aBuf[buf * AELEMS + (rt * 16 + mrow) * ASTR + kk + koff];
                acc[rt] = __builtin_amdgcn_wmma_f32_16x16x4_f32(
                              /*neg_a=*/false, a, /*neg_b=*/false, b,
                              /*c_mod=*/(short)0, acc[rt],
                              /*reuse_a=*/false, /*reuse_b=*/false);
            }
        }

        __syncthreads();                      // all waves done reading aBuf[buf^1]
        if (c + 1 < NCHUNK) {
            #pragma unroll
            for (int j = 0; j < RT; ++j)
                *(float4*)&aBuf[(buf ^ 1) * AELEMS + (sm + 16 * j) * ASTR + sf * 4] = stage[j];
        }
        __syncthreads();                      // stores visible for next iteration
    }
    // trailing barrier above also fences the aBuf -> resb aliasing

    // --- add bias, spill 16x16 f32 tiles to LDS result buffer ---
    {
        const float bias = (mat ? bn : br)[e0 + mrow];
        #pragma unroll
        for (int rt = 0; rt < RT; ++rt) {
            #pragma unroll
            for (int r = 0; r < 8; ++r) {
                const int M = rt * 16 + r + (lane >> 4) * 8;   // C/D layout
                resb[mat * RELEMS + M * RSTR + e0 + mrow] = acc[rt][r] + bias;
            }
        }
    }
    __syncthreads();

    // --- epilogue: noisy logits, top-2, 2-way softmax (one thread per token) ---
    if (t < ROWS) {
        const int tok = tok0 + t;
        float v1 = -__builtin_inff(), v2 = -__builtin_inff();
        int   i1 = 0, i2 = 0;
        for (int e = 0; e < EDIM; ++e) {
            const float lg = resb[t * RSTR + e];
            const float nl = resb[RELEMS + t * RSTR + e];
            // softplus(x) = max(x,0) + log1p(exp(-|x|)) (numerically stable)
            const float sp = fmaxf(nl, 0.f) + log1pf(__expf(-fabsf(nl)));
            const float nv = lg + noise[(size_t)tok * EDIM + e] * sp;
            if (nv > v1)      { v2 = v1; i2 = i1; v1 = nv; i1 = e; }
            else if (nv > v2) { v2 = nv; i2 = e; }
        }
        const float ex  = __expf(v2 - v1);    // softmax over {v1,v2}; others exp(-inf)=0
        const float inv = 1.f / (1.f + ex);
        tp1[t] = inv;
        tp2[t] = ex * inv;
        tix1[t] = i1;
        tix2[t] = i2;
        indices[(size_t)tok * 2 + 0] = i1;
        indices[(size_t)tok * 2 + 1] = i2;
    }
    __syncthreads();

    // --- all 256 threads write coalesced probs (4x float4 each) ---
    #pragma unroll
    for (int j = 0; j < RT; ++j) {
        const int   m  = sm + 16 * j;
        const float p1 = tp1[m], p2 = tp2[m];
        const int   i1 = tix1[m], i2 = tix2[m];
        const int   e  = sf * 4;
        float4 o;
        o.x = (e + 0 == i1) ? p1 : ((e + 0 == i2) ? p2 : 0.f);
        o.y = (e + 1 == i1) ? p1 : ((e + 1 == i2) ? p2 : 0.f);
        o.z = (e + 2 == i1) ? p1 : ((e + 2 == i2) ? p2 : 0.f);
        o.w = (e + 3 == i1) ? p1 : ((e + 3 == i2) ? p2 : 0.f);
        *(float4*)(probs + (size_t)(tok0 + m) * EDIM + e) = o;
    }
}

extern "C" void kernel_launch(void* const* d_in, const int* in_sizes, int n_in,
                              void* d_out, int out_size, void* d_ws, size_t ws_size,
                              hipStream_t stream) {
    const float* X     = (const float*)d_in[0];  // mh_output [4,4096,2048]
    const float* Wr    = (const float*)d_in[1];  // W_route   [2048,64]
    const float* br    = (const float*)d_in[2];  // b_route   [64]
    const float* Wn    = (const float*)d_in[3];  // W_noise   [2048,64]
    const float* bn    = (const float*)d_in[4];  // b_noise   [64]
    const float* noise = (const float*)d_in[5];  // noise     [4,4096,64]

    float* probs   = (float*)d_out;                                   // [4,4096,64] f32
    int*   indices = (int*)((float*)d_out + (size_t)N_TOK * EDIM);    // [4,4096,2] i32

    noisy_topk_router_kernel<<<N_TOK / ROWS, 256, 0, stream>>>(
        X, Wr, br, Wn, bn, noise, probs, indices);
}